// WhiteSoftmax_28406913696444
// MI455X (gfx1250) — compile-verified
//
#include <hip/hip_runtime.h>
#include <hip/hip_bf16.h>
#include <cstdint>

// Problem constants from the reference:
//   x:            (B=8192, C=64, O=128) int32, values in [0,256)
//   add_table:    (128, 256, 256) int32  -> 32 MB, L2-resident
//   softmax_table:(256, 256) float32     -> 256 KB, L2-resident
//   out:          (8192, 64, 128) float32
// chains = B*C = 524288, each runs a serial 128-step FSM over add_table,
// then 128 independent gathers from softmax_table.

#define O_LEN          128
#define CHAINS_PER_BLK 256          // threads per block == chains per block
#define TILE_J         16           // j-values staged per tile
#define N_TILES        (O_LEN / TILE_J)   // 8
#define LDS_STRIDE     20           // dwords per chain per tile: 16 data + 4 pad
                                    // (keeps 16B alignment for b128 async loads,
                                    //  breaks LDS bank conflicts)

__global__ __launch_bounds__(CHAINS_PER_BLK)
void fsm_softmax_kernel(const int* __restrict__ xg,
                        const int* __restrict__ add_tab,
                        const float* __restrict__ soft_tab,
                        float* __restrict__ outg)
{
    // double-buffered x tiles: 2 * 256 chains * 20 dwords * 4B = 40 KB
    __shared__ int sx[2][CHAINS_PER_BLK * LDS_STRIDE];

    const int tid = threadIdx.x;
    const long long chain0 = (long long)blockIdx.x * CHAINS_PER_BLK;
    const long long chain  = chain0 + tid;
    const int lbase = tid * LDS_STRIDE;

    float* orow = outg + chain * O_LEN;

    // Warm a few softmax_table lines into L2 (global_prefetch_b8).
    __builtin_prefetch(soft_tab + ((unsigned)(tid * 257) & 0xFFFFu), 0, 1);

    // Stage x[chain0..chain0+255][j0..j0+15] into sx[buf] using CDNA5
    // async global->LDS copies (tracked by ASYNCcnt, not LOADcnt).
    // 256 chains * 16 ints = 16 KB = 1024 x 16B transfers = 4 per thread.
    auto stage = [&](int t, int buf) {
        const int j0 = t * TILE_J;
#pragma unroll
        for (int r = 0; r < 4; ++r) {
            const int i  = tid + CHAINS_PER_BLK * r;  // 0..1023
            const int cc = i >> 2;                    // chain in block
            const int q  = i & 3;                     // 16B quad within tile
            const int* src = xg + (chain0 + cc) * O_LEN + j0 + q * 4;
            // low 32 bits of a generic LDS pointer == LDS byte offset
            const unsigned lds = (unsigned)(uintptr_t)&sx[buf][cc * LDS_STRIDE + q * 4];
            const unsigned long long ga = (unsigned long long)(uintptr_t)src;
            asm volatile("global_load_async_to_lds_b128 %0, %1, off"
                         :: "v"(lds), "v"(ga) : "memory");
        }
    };

    // ---------------- FSM phase (serial dependent gathers) ----------------
    stage(0, 0);
    asm volatile("s_wait_asynccnt 0x0" ::: "memory");
    __syncthreads();

    int s = 0;
    for (int t = 0; t < N_TILES; ++t) {
        const int buf = t & 1;
        if (t + 1 < N_TILES) stage(t + 1, buf ^ 1);   // prefetch next tile

        const int jhi = t * TILE_J;
#pragma unroll
        for (int jj = 0; jj < TILE_J; ++jj) {
            const int xv = sx[buf][lbase + jj];       // ds_load (hoistable)
            // add_table[j][s][x] : flat dword index = (j<<16)|(s<<8)|x
            s = add_tab[((jhi + jj) << 16) + (s << 8) + xv];  // dependent L2 gather
        }

        asm volatile("s_wait_asynccnt 0x0" ::: "memory");
        __syncthreads();
    }

    // ---------------- Output phase (independent gathers) ----------------
    const float* srow = soft_tab + (s << 8);

    stage(0, 0);
    asm volatile("s_wait_asynccnt 0x0" ::: "memory");
    __syncthreads();

    for (int t = 0; t < N_TILES; ++t) {
        const int buf = t & 1;
        if (t + 1 < N_TILES) stage(t + 1, buf ^ 1);

#pragma unroll
        for (int q = 0; q < TILE_J / 4; ++q) {
            float4 v;
            v.x = srow[sx[buf][lbase + q * 4 + 0]];
            v.y = srow[sx[buf][lbase + q * 4 + 1]];
            v.z = srow[sx[buf][lbase + q * 4 + 2]];
            v.w = srow[sx[buf][lbase + q * 4 + 3]];
            // 16 contiguous bytes per lane -> global_store_b128, L2 write-combines
            *reinterpret_cast<float4*>(orow + t * TILE_J + q * 4) = v;
        }

        asm volatile("s_wait_asynccnt 0x0" ::: "memory");
        __syncthreads();
    }
}

extern "C" void kernel_launch(void* const* d_in, const int* in_sizes, int n_in,
                              void* d_out, int out_size, void* d_ws, size_t ws_size,
                              hipStream_t stream) {
    const int*   x        = (const int*)d_in[0];     // (B, C, O) int32
    const int*   add_tab  = (const int*)d_in[1];     // (O, 256, 256) int32
    const float* soft_tab = (const float*)d_in[2];   // (256, 256) float32
    float*       out      = (float*)d_out;           // (B, C, O) float32

    const long long chains = (long long)in_sizes[0] / O_LEN;   // B*C = 524288
    const int blocks = (int)(chains / CHAINS_PER_BLK);          // 2048

    fsm_softmax_kernel<<<blocks, CHAINS_PER_BLK, 0, stream>>>(x, add_tab, soft_tab, out);
}